// RNN_3126736191672
// MI455X (gfx1250) — compile-verified
//
#include <hip/hip_runtime.h>
#include <stdint.h>

// RNN teacher-forced rollout for MI455X (gfx1250), fp32 WMMA path.
//
//  x_{t+1} = (1-ALPHA) * x_gtf + ALPHA * (x_gtf @ W^T)
//  x_gtf[:, :512] = BETA * teacher[:,:,t] + (1-BETA) * x[:, :512]
//
// One kernel launch per timestep (the device-wide dependency forces a global
// sync; graph replay amortizes launch cost). Each WG owns 16 output neurons,
// stages its 16x2048 W slice + the forced state into LDS, and 8 waves split
// K=2048 into 256-wide slices of V_WMMA_F32_16X16X4_F32 (fp32 everywhere to
// match reference numerics over the 512-step chain).
//
// CDNA5 specifics used:
//  - v_wmma_f32_16x16x4_f32 (fp32 matrix path, ISA 7.12 layouts)
//  - global_load_async_to_lds_b128 + s_wait_asynccnt (async DMA staging,
//    no VGPR round-trip; ISA ch.10 async load/store, ASYNCcnt)
//  - LDS row pitch 2052 words -> bank advance of 4 per row -> conflict-free
//    ds_load_b64 fragment reads across both 16-lane halves.

#define NN      2048
#define NOBS    512
#define BSZ     16
#define TSTEPS  512
#define ALPHA_C 0.01f
#define BETA_C  0.125f
#define XPITCH  2052          // padded row pitch (words)
#define NWAVES  8
#define KSLICE  (NN / NWAVES) // 256

typedef __attribute__((ext_vector_type(2))) float v2f;
typedef __attribute__((ext_vector_type(8))) float v8f;

// Async DMA: 16 bytes per lane, global (saddr + voffset) -> the LDS byte address in lds_byte.
// GVS addressing: 64-bit SGPR base + 32-bit VGPR byte offset.
__device__ __forceinline__ void async_copy_b128(uint32_t lds_byte,
                                                const void* gbase,
                                                uint32_t goff_byte) {
  asm volatile("global_load_async_to_lds_b128 %0, %1, %2"
               :
               : "v"(lds_byte), "v"(goff_byte), "s"(gbase)
               : "memory");
}

__device__ __forceinline__ void wait_async_zero() {
  asm volatile("s_wait_asynccnt 0" ::: "memory");
}

__global__ __launch_bounds__(256) void rnn_init(const float* __restrict__ teacher,
                                                float* __restrict__ x0,
                                                float* __restrict__ out) {
  int gid = blockIdx.x * 256 + threadIdx.x;   // 0 .. BSZ*NN-1
  int b = gid >> 11;
  int k = gid & (NN - 1);
  float v = 0.0f;
  if (k < NOBS) v = teacher[b * (NOBS * TSTEPS) + k * TSTEPS + 0];
  x0[gid] = v;
  out[(size_t)b * NN * TSTEPS + (size_t)k * TSTEPS + 0] = v;
}

__global__ __launch_bounds__(256) void rnn_step(const float* __restrict__ x_in,
                                                float* __restrict__ x_out,
                                                const float* __restrict__ w,
                                                const float* __restrict__ teacher,
                                                float* __restrict__ out,
                                                int t_out, int forced) {
  extern __shared__ float smem[];
  float* xs  = smem;                      // BSZ x XPITCH  (teacher-forced state)
  float* wl  = smem + BSZ * XPITCH;       // 16  x XPITCH  (W row slice for this WG)
  float* red = wl + 16 * XPITCH;          // NWAVES x 256  (cross-wave K reduction)

  const int tid = threadIdx.x;
  const int n0  = blockIdx.x * 16;        // first output neuron owned by this WG

  // Wave-relative LDS byte addresses (low 32 bits of the generic pointer are
  // the AS3 offset per the flat->LDS aperture mapping).
  const uint32_t xs_b = (uint32_t)(uintptr_t)smem;
  const uint32_t wl_b = xs_b + (uint32_t)(BSZ * XPITCH * 4);

  // ---- Async-stage W slice rows [n0, n0+16): contiguous 128 KB from L2 ----
  {
    const float* wbase = w + (size_t)n0 * NN;
#pragma unroll
    for (int it = 0; it < 32; ++it) {
      uint32_t e = (uint32_t)(tid + it * 256) * 4u;  // element idx, multiple of 4
      uint32_t r = e >> 11;
      uint32_t k = e & (NN - 1);
      async_copy_b128(wl_b + (r * XPITCH + k) * 4u, wbase, e * 4u);
    }
  }
  // ---- Async-stage state columns [NOBS, NN) (no forcing applied there) ----
  {
#pragma unroll
    for (int it = 0; it < 24; ++it) {
      uint32_t e  = (uint32_t)(tid + it * 256) * 4u; // 0..24572, multiple of 4
      uint32_t r  = e / (NN - NOBS);                 // row (1536 elems per row)
      uint32_t km = e - r * (NN - NOBS);             // multiple of 4
      async_copy_b128(xs_b + (r * XPITCH + NOBS + km) * 4u,
                      x_in, (r * NN + NOBS + km) * 4u);
    }
  }
  // ---- Forced columns [0, NOBS): blend through VGPRs (teacher is L2-hot) ----
  for (int idx = tid; idx < BSZ * NOBS; idx += 256) {
    int b = idx >> 9;            // NOBS = 512
    int k = idx & (NOBS - 1);
    float v = x_in[b * NN + k];
    if (forced)
      v = BETA_C * teacher[b * (NOBS * TSTEPS) + k * TSTEPS + t_out] + (1.0f - BETA_C) * v;
    xs[b * XPITCH + k] = v;
  }
  wait_async_zero();
  __syncthreads();

  // ---- fp32 WMMA: D(16x16) += A(16x4) x B(4x16), K split across 8 waves ----
  // A layout (ISA 7.12.2): lane<16: {x[m][k], x[m][k+1]}, lane>=16: {x[m][k+2], x[m][k+3]}
  // B layout mirrors it with rows = neurons: same +koff addressing on the W slice.
  const int wave = tid >> 5;
  const int lane = tid & 31;
  const int row  = lane & 15;
  const int koff = (lane >> 4) * 2;
  const float* xrow = &xs[row * XPITCH + koff];
  const float* wrow = &wl[row * XPITCH + koff];

  v8f acc = {};
  const int kbase = wave * KSLICE;
#pragma unroll 8
  for (int k = kbase; k < kbase + KSLICE; k += 4) {
    v2f a = *(const v2f*)(xrow + k);
    v2f bfrag = *(const v2f*)(wrow + k);
    acc = __builtin_amdgcn_wmma_f32_16x16x4_f32(false, a, false, bfrag,
                                                (short)0, acc, false, false);
  }

  // ---- Park partial tile: element (r, lane) is (M = r + (lane>>4)*8, N = lane&15) ----
#pragma unroll
  for (int r = 0; r < 8; ++r) {
    int m = r + (lane >> 4) * 8;
    red[wave * 256 + m * 16 + (lane & 15)] = acc[r];
  }
  __syncthreads();

  // ---- Reduce 8 waves, apply elementwise model, store state + output column ----
  {
    int b  = tid >> 4;       // batch (M)
    int nl = tid & 15;       // local neuron (N)
    float sum = 0.0f;
#pragma unroll
    for (int wv = 0; wv < NWAVES; ++wv) sum += red[wv * 256 + tid];
    float xg   = xs[b * XPITCH + n0 + nl];
    float xnew = (1.0f - ALPHA_C) * xg + ALPHA_C * sum;
    x_out[b * NN + n0 + nl] = xnew;
    if (t_out >= 0)
      out[(size_t)b * NN * TSTEPS + (size_t)(n0 + nl) * TSTEPS + t_out] = xnew;
  }
}

extern "C" void kernel_launch(void* const* d_in, const int* in_sizes, int n_in,
                              void* d_out, int out_size, void* d_ws, size_t ws_size,
                              hipStream_t stream) {
  const float* teacher = (const float*)d_in[0];   // (16, 512, 512) f32
  const float* w       = (const float*)d_in[1];   // (2048, 2048)  f32
  float* out = (float*)d_out;                     // (16, 2048, 512) f32

  float* xA = (float*)d_ws;                       // 16x2048 f32 state ping
  float* xB = xA + BSZ * NN;                      // 16x2048 f32 state pong

  const size_t shbytes =
      (size_t)(BSZ * XPITCH + 16 * XPITCH + NWAVES * 256) * sizeof(float); // ~265 KB < 320 KB

  // out[:,:,0] = x0 ; state <- x0
  rnn_init<<<(BSZ * NN) / 256, 256, 0, stream>>>(teacher, xA, out);

  // x1 = fmodel(x0): unforced, not emitted
  rnn_step<<<NN / 16, 256, shbytes, stream>>>(xA, xB, w, teacher, out, /*t_out=*/-1, /*forced=*/0);

  // 511 forced steps: teacher column t, output column t
  float* cur = xB;
  float* nxt = xA;
  for (int t = 1; t < TSTEPS; ++t) {
    rnn_step<<<NN / 16, 256, shbytes, stream>>>(cur, nxt, w, teacher, out, t, 1);
    float* tmp = cur; cur = nxt; nxt = tmp;
  }
}